// SampledSofmax_12515534700714
// MI455X (gfx1250) — compile-verified
//
#include <hip/hip_runtime.h>
#include <hip/hip_bf16.h>

typedef float v2f __attribute__((ext_vector_type(2)));
typedef float v8f __attribute__((ext_vector_type(8)));

#define UNITS 50257
#define NEGS  8192
#define CDIM  2048
#define NROWS 2048   // B*S = 2*1024

// fragment-order LDS strides (floats)
#define A_HL   160          // half stride: 64 m * 2 + 32 pad  -> bank offset 32
#define A_KG   328          // kgroup stride: >=320, mod 64 == 8 (store conflict-free)
#define A_SLAB (8 * A_KG)   // 2624 floats
#define B_HL   288          // 128 n * 2 + 32 pad -> bank offset 32
#define B_KG   576          // 2 * B_HL
#define B_SLAB (8 * B_KG)   // 4608 floats
#define NSLAB  (CDIM / 32)  // 64

// ---- workspace layout (floats) ----
#define WS_SAMP   ((size_t)0)                        // NROWS*NEGS
#define WS_ADJ    ((size_t)NROWS * NEGS)             // NEGS
#define WS_TRUE   (WS_ADJ + NEGS)                    // NROWS
#define WS_RMAX   (WS_TRUE + NROWS)                  // NROWS
#define WS_RSUM   (WS_RMAX + NROWS)                  // NROWS
#define WS_PEREX  (WS_RSUM + NROWS)                  // NROWS

__device__ __forceinline__ float log_expected_count(int id) {
    // p = (log(id+2)-log(id+1)) / log(UNITS+1) ; E = -expm1(NEG*log1p(-p))
    float f = (float)id;
    float p = (logf(f + 2.0f) - logf(f + 1.0f)) / logf((float)(UNITS + 1));
    float ec = -expm1f((float)NEGS * log1pf(-p));
    return logf(ec);
}

__device__ __forceinline__ void lse_merge(float& m, float& s, float m2, float s2) {
    float M = fmaxf(m, m2);
    s = s * expf(m - M) + s2 * expf(m2 - M);
    m = M;
}

// ============================================================
// WMMA fp32 GEMM core: 256 thr (8 waves), block tile 64(M) x 128(N),
// K-slab 32, double-buffered fragment-order LDS, register prefetch.
// Wave (wm 0..3, wn 0..1) owns 16 x 64 via 4 accumulators.
// COLSRC: 0 = dense columns (clamped at UNITS), 1 = gathered via scol[]
// ============================================================
template <int COLSRC>
__device__ __forceinline__ void gemm_core(const float* __restrict__ X,
                                          const float* __restrict__ W,
                                          const int* __restrict__ scol, // LDS, 128 ids (COLSRC==1)
                                          int tileM, int tileN,
                                          v8f acc[4]) {
    __shared__ float Af[2 * A_SLAB];
    __shared__ float Bf[2 * B_SLAB];

    const int tid  = threadIdx.x;
    const int lane = tid & 31;
    const int wave = tid >> 5;
    const int wm   = wave >> 1;      // 0..3
    const int wn   = wave & 1;       // 0..1
    const int hl   = lane >> 4;
    const int l    = lane & 15;

    // ---- staging geometry ----
    // A: thread loads 2 float4 (rows ar, ar+32 ; k-group akg)
    const int ar  = tid >> 3;        // 0..31
    const int akg = tid & 7;         // 0..7  (covers k = akg*4 .. akg*4+3)
    // B: thread loads 16 scalars (col bc ; rows br0, br0+2, ...)
    const int bc  = tid & 127;       // 0..127
    const int br0 = tid >> 7;        // 0..1
    int colc;
    if (COLSRC == 1) colc = scol[bc];
    else {
        int c = tileN + bc;
        colc = c < UNITS ? c : (UNITS - 1);   // clamp; OOB columns never stored
    }

    // ---- fragment read bases ----
    const int abase = hl * A_HL + (wm * 16 + l) * 2;
    const int bbase = hl * B_HL + (wn * 64 + l) * 2;

    // ---- prologue: fetch + stage slab 0 ----
    float4 a0 = *(const float4*)&X[(size_t)(tileM + ar) * CDIM + akg * 4];
    float4 a1 = *(const float4*)&X[(size_t)(tileM + ar + 32) * CDIM + akg * 4];
    float breg[16];
    #pragma unroll
    for (int i = 0; i < 16; i++) {
        int rr = br0 + 2 * i;
        breg[i] = W[(size_t)rr * UNITS + colc];
    }
    {
        float* A = Af;
        *(v2f*)&A[akg * A_KG + ar * 2]                 = (v2f){a0.x, a0.y};
        *(v2f*)&A[akg * A_KG + A_HL + ar * 2]          = (v2f){a0.z, a0.w};
        *(v2f*)&A[akg * A_KG + (ar + 32) * 2]          = (v2f){a1.x, a1.y};
        *(v2f*)&A[akg * A_KG + A_HL + (ar + 32) * 2]   = (v2f){a1.z, a1.w};
        float* B = Bf;
        #pragma unroll
        for (int i = 0; i < 16; i++) {
            int rr = br0 + 2 * i;                       // e = rr&1, hl = (rr>>1)&1, kg = rr>>2
            B[(rr >> 2) * B_KG + ((rr >> 1) & 1) * B_HL + bc * 2 + (rr & 1)] = breg[i];
        }
    }
    __syncthreads();

    for (int s = 0; s < NSLAB; ++s) {
        const int cur = s & 1;
        const bool more = (s + 1) < NSLAB;
        // ---- prefetch next slab into registers (overlaps WMMA compute) ----
        if (more) {
            const int kn = (s + 1) * 32;
            a0 = *(const float4*)&X[(size_t)(tileM + ar) * CDIM + kn + akg * 4];
            a1 = *(const float4*)&X[(size_t)(tileM + ar + 32) * CDIM + kn + akg * 4];
            #pragma unroll
            for (int i = 0; i < 16; i++) {
                int rr = br0 + 2 * i;
                breg[i] = W[(size_t)(kn + rr) * UNITS + colc];
            }
        }
        // ---- compute: 8 kgroups x (1 A frag + 4 B frags -> 4 WMMA) ----
        {
            const float* A = &Af[cur * A_SLAB + abase];
            const float* B = &Bf[cur * B_SLAB + bbase];
            #pragma unroll
            for (int kg = 0; kg < 8; kg++) {
                v2f a  = *(const v2f*)&A[kg * A_KG];
                v2f b0 = *(const v2f*)&B[kg * B_KG];
                v2f b1 = *(const v2f*)&B[kg * B_KG + 32];
                v2f b2 = *(const v2f*)&B[kg * B_KG + 64];
                v2f b3 = *(const v2f*)&B[kg * B_KG + 96];
                acc[0] = __builtin_amdgcn_wmma_f32_16x16x4_f32(false, a, false, b0,
                                                               (short)0, acc[0], false, false);
                acc[1] = __builtin_amdgcn_wmma_f32_16x16x4_f32(false, a, false, b1,
                                                               (short)0, acc[1], false, false);
                acc[2] = __builtin_amdgcn_wmma_f32_16x16x4_f32(false, a, false, b2,
                                                               (short)0, acc[2], false, false);
                acc[3] = __builtin_amdgcn_wmma_f32_16x16x4_f32(false, a, false, b3,
                                                               (short)0, acc[3], false, false);
            }
        }
        // ---- stage next slab into alternate buffer ----
        if (more) {
            float* A = &Af[(cur ^ 1) * A_SLAB];
            *(v2f*)&A[akg * A_KG + ar * 2]               = (v2f){a0.x, a0.y};
            *(v2f*)&A[akg * A_KG + A_HL + ar * 2]        = (v2f){a0.z, a0.w};
            *(v2f*)&A[akg * A_KG + (ar + 32) * 2]        = (v2f){a1.x, a1.y};
            *(v2f*)&A[akg * A_KG + A_HL + (ar + 32) * 2] = (v2f){a1.z, a1.w};
            float* B = &Bf[(cur ^ 1) * B_SLAB];
            #pragma unroll
            for (int i = 0; i < 16; i++) {
                int rr = br0 + 2 * i;
                B[(rr >> 2) * B_KG + ((rr >> 1) & 1) * B_HL + bc * 2 + (rr & 1)] = breg[i];
            }
        }
        __syncthreads();
    }
}

// ============================================================
// K1: full logits = X @ W + bias   -> d_out (raw logits)
// ============================================================
__global__ void k_gemm_logits(const float* __restrict__ X,
                              const float* __restrict__ W,
                              const float* __restrict__ bias,
                              float* __restrict__ out) {
    const int tileN = blockIdx.x * 128;
    const int tileM = blockIdx.y * 64;
    const int lane  = threadIdx.x & 31;
    const int wave  = threadIdx.x >> 5;
    const int wm = wave >> 1, wn = wave & 1, hl = lane >> 4, l = lane & 15;

    v8f acc[4] = {};
    gemm_core<0>(X, W, nullptr, tileM, tileN, acc);

    #pragma unroll
    for (int j = 0; j < 4; j++) {
        int n = tileN + wn * 64 + j * 16 + l;
        if (n < UNITS) {
            float bv = bias[n];
            #pragma unroll
            for (int r = 0; r < 8; r++) {
                int m = tileM + wm * 16 + r + 8 * hl;
                out[(size_t)m * UNITS + n] = acc[j][r] + bv;
            }
        }
    }
}

// ============================================================
// K5: sampled logits = X @ W[:, sampled] + adj -> ws samp buffer
// ============================================================
__global__ void k_gemm_sampled(const float* __restrict__ X,
                               const float* __restrict__ W,
                               const int* __restrict__ sampled,
                               const float* __restrict__ adj,
                               float* __restrict__ samp) {
    __shared__ int scol[128];
    const int tileN = blockIdx.x * 128;   // NEGS multiple of 128
    const int tileM = blockIdx.y * 64;
    const int lane  = threadIdx.x & 31;
    const int wave  = threadIdx.x >> 5;
    const int wm = wave >> 1, wn = wave & 1, hl = lane >> 4, l = lane & 15;

    if (threadIdx.x < 128) scol[threadIdx.x] = sampled[tileN + threadIdx.x];
    __syncthreads();

    v8f acc[4] = {};
    gemm_core<1>(X, W, scol, tileM, tileN, acc);

    #pragma unroll
    for (int j = 0; j < 4; j++) {
        int n = tileN + wn * 64 + j * 16 + l;
        float av = adj[n];
        #pragma unroll
        for (int r = 0; r < 8; r++) {
            int m = tileM + wm * 16 + r + 8 * hl;
            samp[(size_t)m * NEGS + n] = acc[j][r] + av;
        }
    }
}

// ============================================================
// K2: per-row online (max, sum exp) over UNITS logits
// ============================================================
__global__ void k_row_stats(const float* __restrict__ logits,
                            float* __restrict__ rmax, float* __restrict__ rsum) {
    __shared__ float sm[8], ss[8];
    const int row = blockIdx.x;
    const float* lp = logits + (size_t)row * UNITS;
    float m = -3.4e38f, s = 0.0f;
    for (int i = threadIdx.x; i < UNITS; i += blockDim.x)
        lse_merge(m, s, lp[i], 1.0f);
    for (int off = 16; off > 0; off >>= 1) {
        float m2 = __shfl_xor(m, off, 32);
        float s2 = __shfl_xor(s, off, 32);
        lse_merge(m, s, m2, s2);
    }
    if ((threadIdx.x & 31) == 0) { sm[threadIdx.x >> 5] = m; ss[threadIdx.x >> 5] = s; }
    __syncthreads();
    if (threadIdx.x == 0) {
        float M = sm[0], S = ss[0];
        for (int w = 1; w < 8; w++) lse_merge(M, S, sm[w], ss[w]);
        rmax[row] = M;
        rsum[row] = S;
    }
}

// ============================================================
// K3: in-place softmax normalize
// ============================================================
__global__ void k_normalize(float* __restrict__ out,
                            const float* __restrict__ rmax,
                            const float* __restrict__ rsum) {
    const int row = blockIdx.y;
    const float M = rmax[row];
    const float inv = 1.0f / rsum[row];
    float* p = out + (size_t)row * UNITS;
    int base = blockIdx.x * 1024 + threadIdx.x;
    #pragma unroll
    for (int j = 0; j < 4; j++) {
        int i = base + j * 256;
        if (i < UNITS) p[i] = expf(p[i] - M) * inv;
    }
}

// ============================================================
// K4: per-sampled-column adjustment = bias[s] - log(expected_count)
// ============================================================
__global__ void k_adj(const int* __restrict__ sampled,
                      const float* __restrict__ bias, float* __restrict__ adj) {
    int j = blockIdx.x * blockDim.x + threadIdx.x;
    if (j < NEGS) {
        int s = sampled[j];
        adj[j] = bias[s] - log_expected_count(s);
    }
}

// ============================================================
// K6: true logit per row = dot(x, W[:,t]) + bias[t] - log(E[t])
// ============================================================
__global__ void k_true_logit(const float* __restrict__ X,
                             const float* __restrict__ W,
                             const int* __restrict__ targets,
                             const float* __restrict__ bias,
                             float* __restrict__ tl) {
    __shared__ float red[8];
    const int row = blockIdx.x;
    const int t = targets[row];
    const float* x = X + (size_t)row * CDIM;
    float acc = 0.0f;
    for (int k = threadIdx.x; k < CDIM; k += blockDim.x)
        acc += x[k] * W[(size_t)k * UNITS + t];
    for (int off = 16; off > 0; off >>= 1) acc += __shfl_xor(acc, off, 32);
    if ((threadIdx.x & 31) == 0) red[threadIdx.x >> 5] = acc;
    __syncthreads();
    if (threadIdx.x == 0) {
        float s = 0.0f;
        for (int w = 0; w < 8; w++) s += red[w];
        tl[row] = s + bias[t] - log_expected_count(t);
    }
}

// ============================================================
// K7: per-row sampled-softmax term:
//     per_ex = logsumexp([true, samp(masked)]) - true
// ============================================================
__global__ void k_row_loss(const float* __restrict__ samp,
                           const float* __restrict__ tl,
                           const int* __restrict__ sampled,
                           const int* __restrict__ targets,
                           float* __restrict__ per_ex) {
    __shared__ float sm[8], ss[8];
    const int row = blockIdx.x;
    const int t = targets[row];
    const float tlog = tl[row];
    const float* sp = samp + (size_t)row * NEGS;
    float m = -3.4e38f, s = 0.0f;
    if (threadIdx.x == 0) lse_merge(m, s, tlog, 1.0f);   // include true logit once
    for (int j = threadIdx.x; j < NEGS; j += blockDim.x) {
        if (sampled[j] == t) continue;                   // accidental hit -> skipped, adds nothing to the sum
        lse_merge(m, s, sp[j], 1.0f);
    }
    for (int off = 16; off > 0; off >>= 1) {
        float m2 = __shfl_xor(m, off, 32);
        float s2 = __shfl_xor(s, off, 32);
        lse_merge(m, s, m2, s2);
    }
    if ((threadIdx.x & 31) == 0) { sm[threadIdx.x >> 5] = m; ss[threadIdx.x >> 5] = s; }
    __syncthreads();
    if (threadIdx.x == 0) {
        float M = sm[0], S = ss[0];
        for (int w = 1; w < 8; w++) lse_merge(M, S, sm[w], ss[w]);
        per_ex[row] = (M + logf(S)) - tlog;
    }
}

// ============================================================
// K8: deterministic mean over NROWS -> loss scalar
// ============================================================
__global__ void k_final_loss(const float* __restrict__ per_ex, float* __restrict__ out) {
    __shared__ float red[8];
    float acc = 0.0f;
    for (int i = threadIdx.x; i < NROWS; i += blockDim.x) acc += per_ex[i];
    for (int off = 16; off > 0; off >>= 1) acc += __shfl_xor(acc, off, 32);
    if ((threadIdx.x & 31) == 0) red[threadIdx.x >> 5] = acc;
    __syncthreads();
    if (threadIdx.x == 0) {
        float s = 0.0f;
        for (int w = 0; w < 8; w++) s += red[w];
        out[(size_t)NROWS * UNITS] = s / (float)NROWS;
    }
}

extern "C" void kernel_launch(void* const* d_in, const int* in_sizes, int n_in,
                              void* d_out, int out_size, void* d_ws, size_t ws_size,
                              hipStream_t stream) {
    const float* X    = (const float*)d_in[0];   // [2,1024,2048]
    const int*   tgt  = (const int*)  d_in[1];   // [2,1024]
    const float* W    = (const float*)d_in[2];   // [2048,50257]
    const float* bias = (const float*)d_in[3];   // [50257]
    const int*   smp  = (const int*)  d_in[4];   // [8192]
    float* out = (float*)d_out;                  // [2*1024*50257 probs] + [1 loss]
    float* ws  = (float*)d_ws;

    float* samp  = ws + WS_SAMP;
    float* adj   = ws + WS_ADJ;
    float* tl    = ws + WS_TRUE;
    float* rmax  = ws + WS_RMAX;
    float* rsum  = ws + WS_RSUM;
    float* perex = ws + WS_PEREX;

    // 1) full logits GEMM -> out (raw)
    dim3 g1((UNITS + 127) / 128, NROWS / 64);
    k_gemm_logits<<<g1, 256, 0, stream>>>(X, W, bias, out);

    // 2) row stats, 3) normalize in place
    k_row_stats<<<NROWS, 256, 0, stream>>>(out, rmax, rsum);
    dim3 g3((UNITS + 1023) / 1024, NROWS);
    k_normalize<<<g3, 256, 0, stream>>>(out, rmax, rsum);

    // 4) sampled-class adjustments
    k_adj<<<(NEGS + 255) / 256, 256, 0, stream>>>(smp, bias, adj);

    // 5) gathered sampled GEMM -> ws
    dim3 g5(NEGS / 128, NROWS / 64);
    k_gemm_sampled<<<g5, 256, 0, stream>>>(X, W, smp, adj, samp);

    // 6) true logits, 7) per-row loss, 8) mean
    k_true_logit<<<NROWS, 256, 0, stream>>>(X, W, tgt, bias, tl);
    k_row_loss<<<NROWS, 256, 0, stream>>>(samp, tl, smp, tgt, perex);
    k_final_loss<<<1, 256, 0, stream>>>(perex, out);
}